// BlockRelLinear_16587163697547
// MI455X (gfx1250) — compile-verified
//
#include <hip/hip_runtime.h>

#define NRELS 128
#define NB    16
#define FEAT  128           // in_features == out_features == 128
#define WPT   1024          // weights per relation = 16 blocks * 8 * 8

typedef float v2f __attribute__((ext_vector_type(2)));
typedef float v4f __attribute__((ext_vector_type(4)));
typedef float v8f __attribute__((ext_vector_type(8)));

// ---------------- counting sort by relation ----------------

__global__ void brl_zero(int* counts) {
  counts[threadIdx.x] = 0;                       // 128 threads
}

__global__ void brl_hist(const int* __restrict__ rel, int* __restrict__ counts, int P) {
  int p = blockIdx.x * blockDim.x + threadIdx.x;
  if (p < P) atomicAdd(&counts[rel[p]], 1);
}

__global__ void brl_scan(const int* __restrict__ counts, int* __restrict__ offs,
                         int* __restrict__ cursors, int* __restrict__ tileoffs,
                         int* __restrict__ totalTiles) {
  if (threadIdx.x == 0) {
    int s = 0, t = 0;
    for (int r = 0; r < NRELS; ++r) {
      offs[r] = s; cursors[r] = s; tileoffs[r] = t;
      s += counts[r];
      t += (counts[r] + 15) >> 4;
    }
    tileoffs[NRELS] = t;
    *totalTiles = t;
  }
}

__global__ void brl_scatter(const int* __restrict__ rel, int* __restrict__ cursors,
                            int* __restrict__ sorted, int P) {
  int p = blockIdx.x * blockDim.x + threadIdx.x;
  if (p < P) {
    int pos = atomicAdd(&cursors[rel[p]], 1);
    sorted[pos] = p;
  }
}

__global__ void brl_tilemap(const int* __restrict__ tileoffs, int* __restrict__ tile2rel) {
  int r = threadIdx.x;                           // 128 threads
  int t0 = tileoffs[r], t1 = tileoffs[r + 1];
  for (int t = t0; t < t1; ++t) tile2rel[t] = r;
}

// ---------------- WMMA tile kernel: 1 wave = 16 same-relation points ----------------

__launch_bounds__(32)
__global__ void brl_gemm(const float* __restrict__ x, const float* __restrict__ blocks,
                         const int* __restrict__ sorted, const int* __restrict__ offs,
                         const int* __restrict__ counts, const int* __restrict__ tileoffs,
                         const int* __restrict__ tile2rel, const int* __restrict__ totalTiles,
                         float* __restrict__ out) {
  __shared__ float wsm[WPT];        // relation weights [blk][i][o]
  __shared__ float xsm[16 * FEAT];  // 16 gathered input rows
  __shared__ float osm[16 * FEAT];  // 16 output rows (store transpose staging)
  __shared__ int   pidx[16];

  const int bid = blockIdx.x;
  if (bid >= *totalTiles) return;                // uniform across the wave

  const int r    = tile2rel[bid];
  const int tIn  = bid - tileoffs[r];
  const int base = offs[r];
  const int cnt  = counts[r];
  const int mrem = cnt - tIn * 16;
  const int m    = mrem < 16 ? mrem : 16;        // valid points in this tile
  const int lane = threadIdx.x;

  const float* wsrc = blocks + (size_t)r * WPT;
  __builtin_prefetch(wsrc, 0, 1);                // global_prefetch_b8 of 4KB weight block

  // stage weights: 256 float4 with 32 lanes
  {
    const v4f* src = (const v4f*)wsrc;
    v4f* dst = (v4f*)wsm;
#pragma unroll
    for (int it = 0; it < 8; ++it) dst[lane + it * 32] = src[lane + it * 32];
  }
  // stage (clamped) point indices
  if (lane < 16) {
    int g = tIn * 16 + lane;
    pidx[lane] = sorted[base + (g < cnt ? g : cnt - 1)];
  }
  __syncthreads();
  // stage 16 input rows, each a coalesced 512B burst
  {
    v4f* dst = (v4f*)xsm;
#pragma unroll
    for (int row = 0; row < 16; ++row) {
      const v4f* src = (const v4f*)(x + (size_t)pidx[row] * FEAT);
      dst[row * 32 + lane] = src[lane];
    }
  }
  __syncthreads();

  const int oN   = lane & 15;                    // D column / A row (M)
  const int half = (lane < 16) ? 0 : 2;          // K sub-index held by this lane half
  const int rofs = (lane < 16) ? 0 : 8;          // D row offset for this lane half
  const bool oLow = (oN < 8);                    // lane owns low 8 output cols of a pair
  const float* xrow  = &xsm[oN * FEAT];
  const float* wbase = &wsm[oN & 7];

#pragma unroll
  for (int n = 0; n < 8; ++n) {                  // block pair -> out cols [16n, 16n+16)
    const float* wlo = wbase + (2 * n) * 64;     // block 2n   (feeds cols oN<8)
    const float* whi = wlo + 64;                 // block 2n+1 (feeds cols oN>=8)
    v8f acc = {};
#if defined(__has_builtin) && __has_builtin(__builtin_amdgcn_wmma_f32_16x16x4_f32)
#pragma unroll
    for (int j = 0; j < 4; ++j) {
      const int k  = 4 * j + half;               // K within the 16-wide pair
      const int i0 = k & 7;
      v2f a;
      a.x = xrow[16 * n + k];
      a.y = xrow[16 * n + k + 1];
      // block-diagonal B: unconditional LDS loads, value-select only (v_cndmask,
      // no EXEC manipulation). j<2 -> block 2n / low cols; j>=2 -> block 2n+1 / high.
      const float* wsel = (j < 2) ? wlo : whi;   // compile-time under unroll
      const float w0 = wsel[i0 * 8];
      const float w1 = wsel[i0 * 8 + 8];
      const bool  on = (j < 2) ? oLow : !oLow;   // lane-constant mask
      v2f b;
      b.x = on ? w0 : 0.0f;
      b.y = on ? w1 : 0.0f;
      acc = __builtin_amdgcn_wmma_f32_16x16x4_f32(false, a, false, b, (short)0, acc,
                                                  false, false);
    }
#else
    // VALU fallback (also taken by the host pass): direct block-diagonal matvec
#pragma unroll
    for (int v = 0; v < 8; ++v) {
      const int prow = v + rofs;
      const int blk  = 2 * n + (oN >> 3);
      float s = acc[v];
#pragma unroll
      for (int i = 0; i < 8; ++i)
        s += xsm[prow * FEAT + blk * 8 + i] * wsm[blk * 64 + i * 8 + (oN & 7)];
      acc[v] = s;
    }
#endif
#pragma unroll
    for (int v = 0; v < 8; ++v)
      osm[(v + rofs) * FEAT + 16 * n + oN] = acc[v];
  }
  __syncthreads();

  // coalesced 512B row stores (masked for tail tiles)
#pragma unroll
  for (int row = 0; row < 16; ++row) {
    if (row < m) {
      v4f val = *(const v4f*)&osm[row * FEAT + lane * 4];
      *(v4f*)(out + (size_t)pidx[row] * FEAT + lane * 4) = val;
    }
  }
}

// ---------------- launch ----------------

extern "C" void kernel_launch(void* const* d_in, const int* in_sizes, int n_in,
                              void* d_out, int out_size, void* d_ws, size_t ws_size,
                              hipStream_t stream) {
  const float* x      = (const float*)d_in[0];
  const float* blocks = (const float*)d_in[1];
  const int*   rel    = (const int*)d_in[2];
  float* out = (float*)d_out;
  const int P = in_sizes[2];

  int* wsI      = (int*)d_ws;
  int* counts   = wsI;              // 128
  int* offs     = wsI + 128;        // 128
  int* cursors  = wsI + 256;        // 128
  int* tileoffs = wsI + 384;        // 129
  int* totalT   = wsI + 516;        // 1
  int* sorted   = wsI + 1024;       // P
  int* tile2rel = wsI + 1024 + P;   // <= P/16 + 128

  brl_zero<<<1, NRELS, 0, stream>>>(counts);
  brl_hist<<<(P + 255) / 256, 256, 0, stream>>>(rel, counts, P);
  brl_scan<<<1, 32, 0, stream>>>(counts, offs, cursors, tileoffs, totalT);
  brl_scatter<<<(P + 255) / 256, 256, 0, stream>>>(rel, cursors, sorted, P);
  brl_tilemap<<<1, NRELS, 0, stream>>>(tileoffs, tile2rel);

  const int maxTiles = (P + 15) / 16 + NRELS;    // host-side upper bound; device checks totalT
  brl_gemm<<<maxTiles, 32, 0, stream>>>(x, blocks, sorted, offs, counts,
                                        tileoffs, tile2rel, totalT, out);
}